// SpringNN_63625645523421
// MI455X (gfx1250) — compile-verified
//
#include <hip/hip_runtime.h>
#include <math.h>

typedef __attribute__((ext_vector_type(2))) float v2f;
typedef __attribute__((ext_vector_type(4))) float v4f;
typedef __attribute__((ext_vector_type(8))) float v8f;

#define SPR_S 32
#define SPR_H 32
#define TPB 256
#define WAVES_PER_BLOCK (TPB / 32)
#define PTS_PER_TILE 16

// Prefer the gfx1250 hardware tanh (TRANS unit) if the builtin exists.
#if __has_builtin(__builtin_amdgcn_tanhf)
  #define FAST_TANH(x) __builtin_amdgcn_tanhf(x)
#elif __has_builtin(__builtin_amdgcn_tanh_f32)
  #define FAST_TANH(x) __builtin_amdgcn_tanh_f32(x)
#else
  #define FAST_TANH(x) tanhf(x)
#endif

// out[n] = sum_s w[s,n] * ( sum_h tanh(t_n*W1[s,h]+b1[s,h]) * W2[s,h] + b2[s] )
// w[s,n] = cosine window, width 4:  w = 0.5*(1 - cos(pi*(t-lo)/2)) for t in [lo, lo+4]
//
// WMMA mapping (V_WMMA_F32_16X16X4_F32, wave32):
//   M (rows)    = 16 points of this wave's tile
//   K           = 4-wide chunks of the (s,h) axis; 8 chunks per spring, 32 springs
//   A[m,k]      = w[s,m] * tanh(t_m*W1[s,k]+b1[s,k])   (window folded into A)
//   B[k,n]      = W2[s,k]  for every column n  -> all 16 columns of D are identical
//   C/D (v8f)   = running accumulator across all 256 WMMAs
// f32 A 16x4 layout (ISA 7.12.2): lanes 0-15 hold K={0,1} (VGPR0,1), lanes 16-31 K={2,3}.
// B 4x16 mirrors this half-major split, so each lane needs hidden indices
// k = 4j + 2*(lane>>4) + {0,1} for both A and B. D: VGPR r = row r (lanes<16) / 8+r.
__global__ __launch_bounds__(TPB) void springnn_wmma_f32(
    const float* __restrict__ t,
    const float* __restrict__ W1,   // [S][1][H]
    const float* __restrict__ b1,   // [S][H]
    const float* __restrict__ W2,   // [S][H][1]
    const float* __restrict__ b2,   // [S]
    const float* __restrict__ xmin, // [S]
    float* __restrict__ out, int N)
{
  __shared__ float sW1[SPR_S * SPR_H];
  __shared__ float sB1[SPR_S * SPR_H];
  __shared__ float sW2[SPR_S * SPR_H];
  __shared__ float sB2[SPR_S];
  __shared__ float sLo[SPR_S];

  // Stage all weights into LDS once per block (~12.5 KB of the 320 KB/WGP pool).
  for (int i = threadIdx.x; i < SPR_S * SPR_H; i += TPB) {
    sW1[i] = W1[i];
    sB1[i] = b1[i];
    sW2[i] = W2[i];
  }
  if (threadIdx.x < SPR_S) {
    sB2[threadIdx.x] = b2[threadIdx.x];
    sLo[threadIdx.x] = xmin[threadIdx.x];
  }
  __syncthreads();

  const int lane  = threadIdx.x & 31;
  const int wave  = threadIdx.x >> 5;
  const int kh    = lane >> 4;    // K half: 0 -> k%4 in {0,1}, 1 -> k%4 in {2,3}
  const int p     = lane & 15;    // this lane's point within the tile (= A row M)
  const int tile  = blockIdx.x * WAVES_PER_BLOCK + wave;
  const int base  = tile * PTS_PER_TILE;
  if (base >= N) return;          // wave-uniform: EXEC stays all-ones for WMMA

  const int tidx  = base + p;
  const float tv  = t[tidx < N ? tidx : (N - 1)];

  v8f acc = {};                   // D accumulator (all 16 columns identical)
  float accB2 = 0.0f;             // sum_s w[s,p]*b2[s] for this lane's point

  for (int s = 0; s < SPR_S; ++s) {
    const float lo = sLo[s];
    const float d  = tv - lo;     // window width is 4.0 (xmax - xmin)
    // w = (1 + cos(pi*(d/2 - 1)))/2 = (1 - cos(pi*d/2))/2, zero outside [0,4]
    const float cw = 0.5f * (1.0f - __cosf(1.57079632679489662f * d));
    const float w  = (d >= 0.0f && d <= 4.0f) ? cw : 0.0f;
    accB2 = fmaf(w, sB2[s], accB2);

    const float* w1s = &sW1[s * SPR_H];
    const float* b1s = &sB1[s * SPR_H];
    const float* w2s = &sW2[s * SPR_H];

#pragma unroll
    for (int j = 0; j < 8; ++j) {
      const int k = 4 * j + 2 * kh;          // even -> 8B-aligned LDS pair loads
      const v2f w1p = *(const v2f*)(w1s + k);
      const v2f b1p = *(const v2f*)(b1s + k);
      const v2f w2p = *(const v2f*)(w2s + k);
      v2f a;
      a.x = w * FAST_TANH(fmaf(tv, w1p.x, b1p.x));
      a.y = w * FAST_TANH(fmaf(tv, w1p.y, b1p.y));
      // D = A*B + C, full f32: (neg_a, A, neg_b, B, c_mod, C, reuse_a, reuse_b)
      acc = __builtin_amdgcn_wmma_f32_16x16x4_f32(
          false, a, false, w2p, (short)0, acc, false, false);
    }
  }

  // Merge the per-point b2 window term. Point q's accB2 lives in lane q (q<16);
  // lane 0 stores rows 0..7 (col 0), lane 16 stores rows 8..15 -> uniform readlanes.
  const int ib2 = __builtin_bit_cast(int, accB2);
  v8f res;
#pragma unroll
  for (int r = 0; r < 8; ++r) {
    const float addLo = __builtin_bit_cast(float, __builtin_amdgcn_readlane(ib2, r));
    const float addHi = __builtin_bit_cast(float, __builtin_amdgcn_readlane(ib2, r + 8));
    res[r] = acc[r] + (kh ? addHi : addLo);
  }

  if (p == 0) {                   // lanes 0 and 16: 8 contiguous outputs each
    const int o0 = base + kh * 8;
    if (o0 + 8 <= N) {
      v4f lo4 = {res[0], res[1], res[2], res[3]};
      v4f hi4 = {res[4], res[5], res[6], res[7]};
      *(v4f*)(out + o0)     = lo4;
      *(v4f*)(out + o0 + 4) = hi4;
    } else {
      for (int r = 0; r < 8 && o0 + r < N; ++r) out[o0 + r] = res[r];
    }
  }
}

extern "C" void kernel_launch(void* const* d_in, const int* in_sizes, int n_in,
                              void* d_out, int out_size, void* d_ws, size_t ws_size,
                              hipStream_t stream) {
  (void)n_in; (void)d_ws; (void)ws_size; (void)out_size;
  const float* t    = (const float*)d_in[0];
  const float* W1   = (const float*)d_in[1];
  const float* b1   = (const float*)d_in[2];
  const float* W2   = (const float*)d_in[3];
  const float* b2   = (const float*)d_in[4];
  const float* xmin = (const float*)d_in[5];
  // d_in[6] = xmax, unused: width is the fixed 4.0 of the reference.
  const int N = in_sizes[0];

  const int tiles  = (N + PTS_PER_TILE - 1) / PTS_PER_TILE;
  const int blocks = (tiles + WAVES_PER_BLOCK - 1) / WAVES_PER_BLOCK;
  springnn_wmma_f32<<<blocks, TPB, 0, stream>>>(t, W1, b1, W2, b2, xmin,
                                                (float*)d_out, N);
}